// graphConvUnpool_55834574848182
// MI455X (gfx1250) — compile-verified
//
#include <hip/hip_runtime.h>
#include <hip/hip_bf16.h>

// ---------------------------------------------------------------------------
// graphConvUnpool on MI455X (gfx1250, wave32)
//   out = relu( D^-1/2 (A+I) D^-1/2 (unpool(x) @ W) + b )
// Pipeline:
//   k_init      : zero h[N,128], zero out[N,128], deg[N] = 1.0 (self loop)
//   k_degree    : deg[dst] += 1 per edge (global_atomic_add_f32)
//   k_rsqrt     : deg = rsqrt(deg)   (dinv)
//   k_gemm      : h[indices[r]] = (x @ W)[r]  via V_WMMA_F32_16X16X4_F32,
//                 W K-slices DMA'd to LDS by the Tensor Data Mover
//                 (double-buffered, TENSORcnt-synchronized, HW LDS padding)
//   k_edge_agg  : out[dst] += h[src] * dinv[src]*dinv[dst]  (f32 atomics)
//   k_finalize  : out = relu(out + h*dinv^2 + b)            (self loop + bias)
// ---------------------------------------------------------------------------

typedef float v2f __attribute__((ext_vector_type(2)));
typedef float v8f __attribute__((ext_vector_type(8)));
typedef unsigned int v4u __attribute__((ext_vector_type(4)));
typedef int v4i __attribute__((ext_vector_type(4)));
typedef int v8i __attribute__((ext_vector_type(8)));

#define FD 128   // feature dim (fixed by reference)
#define PAD 4    // LDS row padding in floats (== TDM pad_amount of 4 DWORDs)

#if __has_builtin(__builtin_amdgcn_tensor_load_to_lds) && \
    __has_builtin(__builtin_amdgcn_s_wait_tensorcnt)
#define USE_TDM 1
#else
#define USE_TDM 0
#endif

// ---------------------------------------------------------------------------
__global__ __launch_bounds__(256) void k_init(float* __restrict__ h,
                                              float* __restrict__ out,
                                              float* __restrict__ deg,
                                              int n_elems, int n_nodes) {
  int stride = gridDim.x * blockDim.x;
  for (int i = blockIdx.x * blockDim.x + threadIdx.x; i < n_elems; i += stride) {
    h[i] = 0.0f;
    out[i] = 0.0f;
    if (i < n_nodes) deg[i] = 1.0f;  // self loop contributes 1 to degree
  }
}

// ---------------------------------------------------------------------------
__global__ __launch_bounds__(256) void k_degree(const int* __restrict__ dst,
                                                float* __restrict__ deg, int E) {
  int stride = gridDim.x * blockDim.x;
  for (int e = blockIdx.x * blockDim.x + threadIdx.x; e < E; e += stride) {
    atomicAdd(&deg[dst[e]], 1.0f);
  }
}

// ---------------------------------------------------------------------------
__global__ __launch_bounds__(256) void k_rsqrt(float* __restrict__ deg, int n) {
  int i = blockIdx.x * blockDim.x + threadIdx.x;
  if (i < n) deg[i] = rsqrtf(deg[i]);  // deg >= 1 always
}

// ---------------------------------------------------------------------------
#if USE_TDM
// TDM DMA: 32x128 f32 tile of W (row-major, stride 128) -> LDS at lds_off,
// with hardware padding of 4 DWORDs after every 128 DWORDs (row stride 132).
// D# built per CDNA5 ISA 8.3/8.4 (group0: count/lds/global/type=2,
// group1: data_size=4B, dims, tile dims, strides, pad fields).
__device__ __forceinline__ void tdm_load_w_slice(const float* gsrc,
                                                 unsigned lds_off) {
  unsigned long long ga = (unsigned long long)(uintptr_t)gsrc;
  v4u g0;
  g0[0] = 1u;                                         // count=1, user mode
  g0[1] = lds_off;                                    // lds_addr
  g0[2] = (unsigned)(ga & 0xFFFFFFFFu);               // global_addr[31:0]
  g0[3] = (unsigned)((ga >> 32) & 0x01FFFFFFu)        // global_addr[56:32]
        | (2u << 30);                                 // type = 2 ("image")
  v8i g1;
  g1[0] = (2 << 16)                                   // data_size = 4 bytes
        | (1 << 20)                                   // pad_enable
        | (6 << 22)                                   // pad_interval = 128 dw
        | (3 << 25);                                  // pad_amount  = 4 dw
  g1[1] = (FD & 0xFFFF) << 16;                        // tensor_dim0 = 128
  g1[2] = (FD & 0xFFFF) << 16;                        // tensor_dim1 = 128
  g1[3] = (FD & 0xFFFF) << 16;                        // tile_dim0   = 128
  g1[4] = 32;                                         // tile_dim1=32, dim2=0
  g1[5] = FD;                                         // dim0_stride = 128
  g1[6] = (FD * FD & 0xFFFF) << 16;                   // dim1_stride lo16
  g1[7] = 0;
  v4i z = {0, 0, 0, 0};
#if defined(__clang_major__) && __clang_major__ >= 23
  v8i z8 = {0, 0, 0, 0, 0, 0, 0, 0};
  __builtin_amdgcn_tensor_load_to_lds(g0, g1, z, z, z8, 0);
#else
  __builtin_amdgcn_tensor_load_to_lds(g0, g1, z, z, 0);
#endif
}
#endif

// ---------------------------------------------------------------------------
// GEMM + unpool-scatter:  h[indices[r], :] = x[r, :] @ W   for r in [0, K)
// Block = 256 threads = 8 wave32; block computes 64 rows x 128 cols.
// Wave w owns column tile ct = w (16 cols), all 4 row tiles.
// K-dim (=128): 4 LDS slices of 32, double-buffered via TDM, 8 WMMA k-steps.
__global__ __launch_bounds__(256) void k_gemm(const float* __restrict__ x,
                                              const float* __restrict__ W,
                                              const int* __restrict__ idx,
                                              float* __restrict__ h, int K) {
  __shared__ float xs[64][FD + PAD];      // A tile, conflict-padded
  __shared__ float wsh[2][32][FD + PAD];  // double-buffered W K-slices

  const int tid = threadIdx.x;
  const int lane = tid & 31;
  const int rowBase = blockIdx.x * 64;

  // scalar (SGPR) wave id -> s_cbranch, so only wave 0 issues TDM ops
  const bool leader = (__builtin_amdgcn_readfirstlane(tid) < 32);

#if USE_TDM
  if (leader) {
    tdm_load_w_slice(W, (unsigned)(uintptr_t)&wsh[0][0][0]);  // slice 0
  }
#endif

  // ---- stage x tile (64 x 128), zero-fill rows past K ----
  for (int q = tid; q < 64 * (FD / 4); q += 256) {
    int row = q >> 5;          // 0..63
    int c4 = q & 31;           // float4 column
    float4 v = make_float4(0.f, 0.f, 0.f, 0.f);
    int gr = rowBase + row;
    if (gr < K) v = ((const float4*)(x + (size_t)gr * FD))[c4];
    ((float4*)&xs[row][0])[c4] = v;
  }

  const int m = lane & 15;          // M (A) / N (B,C) index within tile
  const int kk = (lane >> 4) << 1;  // lanes 16..31 carry K+2 per ISA layout
  const int ct16 = (tid >> 5) * 16; // this wave's column-tile base

  v8f acc[4];
#pragma unroll
  for (int ti = 0; ti < 4; ++ti) acc[ti] = (v8f)(0.0f);

  for (int ks = 0; ks < 4; ++ks) {
    const int buf = ks & 1;
#if USE_TDM
    if (leader) {
      if (ks < 3) {  // prefetch next slice into the other buffer
        tdm_load_w_slice(W + (size_t)(ks + 1) * 32 * FD,
                         (unsigned)(uintptr_t)&wsh[buf ^ 1][0][0]);
        __builtin_amdgcn_s_wait_tensorcnt(1);  // oldest (this slice) done
      } else {
        __builtin_amdgcn_s_wait_tensorcnt(0);  // last slice done
      }
    }
    __syncthreads();
#else
    __syncthreads();
    for (int q = tid; q < 32 * (FD / 4); q += 256) {
      int row = q >> 5;
      int c4 = q & 31;
      ((float4*)&wsh[buf][row][0])[c4] =
          ((const float4*)(W + (size_t)(ks * 32 + row) * FD))[c4];
    }
    __syncthreads();
#endif

#pragma unroll
    for (int k0 = 0; k0 < 32; k0 += 4) {
      v2f bf;  // B fragment: 4x16 slice of W, shared by all 4 row tiles
      bf.x = wsh[buf][k0 + kk + 0][ct16 + m];
      bf.y = wsh[buf][k0 + kk + 1][ct16 + m];
#pragma unroll
      for (int ti = 0; ti < 4; ++ti) {
        v2f af;  // A fragment: 16x4 slice of x tile
        af.x = xs[ti * 16 + m][ks * 32 + k0 + kk + 0];
        af.y = xs[ti * 16 + m][ks * 32 + k0 + kk + 1];
        acc[ti] = __builtin_amdgcn_wmma_f32_16x16x4_f32(
            false, af, false, bf, (short)0, acc[ti], false, false);
      }
    }
    __syncthreads();  // slice consumed; buffer may be overwritten next iter
  }

  // ---- scatter C tiles through the unpool index map ----
  const int half8 = (lane >> 4) * 8;  // lanes 16..31 hold M = 8..15
#pragma unroll
  for (int ti = 0; ti < 4; ++ti) {
#pragma unroll
    for (int j = 0; j < 8; ++j) {
      int r = rowBase + ti * 16 + half8 + j;
      if (r < K) {
        int gr = idx[r];
        h[(size_t)gr * FD + ct16 + m] = acc[ti][j];
      }
    }
  }
}

// ---------------------------------------------------------------------------
// One wave per edge: 32 lanes x float4 = 128 floats. Scatter-add via f32
// atomics (no return -> STOREcnt path). Prefetch next source row.
__global__ __launch_bounds__(256) void k_edge_agg(const int* __restrict__ src,
                                                  const int* __restrict__ dst,
                                                  const float* __restrict__ dinv,
                                                  const float* __restrict__ h,
                                                  float* __restrict__ out, int E) {
  const int lane = threadIdx.x & 31;
  const int wave = threadIdx.x >> 5;
  const int gwave = blockIdx.x * 8 + wave;
  const int nwaves = gridDim.x * 8;

  for (int e = gwave; e < E; e += nwaves) {
    int en = e + nwaves;
    if (en < E) {  // prefetch next edge's source row (global_prefetch_b8)
      __builtin_prefetch(h + (size_t)src[en] * FD + lane * 4, 0, 1);
    }
    int s = src[e];
    int t = dst[e];
    float norm = dinv[s] * dinv[t];
    float4 v = ((const float4*)(h + (size_t)s * FD))[lane];
    float* o = out + (size_t)t * FD + lane * 4;
    atomicAdd(o + 0, v.x * norm);
    atomicAdd(o + 1, v.y * norm);
    atomicAdd(o + 2, v.z * norm);
    atomicAdd(o + 3, v.w * norm);
  }
}

// ---------------------------------------------------------------------------
// out = relu(out + h * dinv^2 + b)   (folds self-loop message + bias)
__global__ __launch_bounds__(256) void k_finalize(const float* __restrict__ h,
                                                  const float* __restrict__ dinv,
                                                  const float* __restrict__ b,
                                                  float* __restrict__ out,
                                                  int n_elems) {
  int i = blockIdx.x * blockDim.x + threadIdx.x;
  if (i >= n_elems) return;
  int r = i >> 7;        // / 128
  int d = i & (FD - 1);  // % 128
  float di = dinv[r];
  float v = out[i] + h[i] * di * di + b[d];
  out[i] = fmaxf(v, 0.0f);
}

// ---------------------------------------------------------------------------
extern "C" void kernel_launch(void* const* d_in, const int* in_sizes, int n_in,
                              void* d_out, int out_size, void* d_ws, size_t ws_size,
                              hipStream_t stream) {
  // setup_inputs order: x_skip[N,128], e_skip[2,E], indices[K], x[K,128],
  //                     W[128,128], b[128]
  const int* e_skip = (const int*)d_in[1];
  const int* indices = (const int*)d_in[2];
  const float* x = (const float*)d_in[3];
  const float* W = (const float*)d_in[4];
  const float* b = (const float*)d_in[5];

  const int N = in_sizes[0] / FD;
  const int E = in_sizes[1] / 2;
  const int K = in_sizes[2];

  const int* src = e_skip;
  const int* dst = e_skip + E;

  float* out = (float*)d_out;
  float* h = (float*)d_ws;                  // N*128 floats
  float* deg = h + (size_t)N * FD;          // N floats (deg -> dinv in place)

  const int n_elems = N * FD;

  k_init<<<4096, 256, 0, stream>>>(h, out, deg, n_elems, N);
  k_degree<<<4096, 256, 0, stream>>>(dst, deg, E);
  k_rsqrt<<<(N + 255) / 256, 256, 0, stream>>>(deg, N);
  k_gemm<<<(K + 63) / 64, 256, 0, stream>>>(x, W, indices, h, K);
  k_edge_agg<<<8192, 256, 0, stream>>>(src, dst, deg, h, out, E);
  k_finalize<<<(n_elems + 255) / 256, 256, 0, stream>>>(h, deg, b, out, n_elems);
}